// GAT_50586124812836
// MI455X (gfx1250) — compile-verified
//
#include <hip/hip_runtime.h>
#include <math.h>

// ---------------------------------------------------------------------------
// GAT 2-layer forward for MI455X (gfx1250, wave32).
//  * GEMMs: V_WMMA_F32_16X16X4_F32, B-matrix staged into LDS by the Tensor
//    Data Mover (tensor_load_to_lds, TENSORcnt), double-buffered.
//  * Edge softmax: int-keyed atomic max + atomic-add segment sums.
// ---------------------------------------------------------------------------

typedef __attribute__((ext_vector_type(2))) float    v2f;
typedef __attribute__((ext_vector_type(8))) float    v8f;
typedef __attribute__((ext_vector_type(4))) unsigned u32x4;
typedef __attribute__((ext_vector_type(4))) int      i32x4;
typedef __attribute__((ext_vector_type(8))) int      i32x8;

#define FIN 512
#define H1 8
#define D1 8
#define C1 64   // H1*D1
#define D2 32
#define NEG_SLOPE 0.2f

__device__ __forceinline__ float lrelu(float v) { return v > 0.f ? v : NEG_SLOPE * v; }

__device__ __forceinline__ void atomicMaxF(unsigned* addr, float v) {
  unsigned uv = __float_as_uint(v);
  unsigned key = (uv & 0x80000000u) ? ~uv : (uv | 0x80000000u);
  atomicMax(addr, key);
}
__device__ __forceinline__ float decodeMaxKey(unsigned key) {
  unsigned uv = (key & 0x80000000u) ? (key & 0x7FFFFFFFu) : ~key;
  return __uint_as_float(uv);
}

// ---------------------------------------------------------------------------
// TDM: issue an async 1D copy of `elems` f32 from global `src` to LDS `dst`.
// D# group0: count=1 | lds_addr | global_addr[56:0] | type=2.
// D# group1: data_size=4B, tensor_dim0=tile_dim0=elems, tensor_dim1=1.
// ---------------------------------------------------------------------------
#if __has_builtin(__builtin_amdgcn_tensor_load_to_lds)
#define HAVE_TDM 1
__device__ __forceinline__ void tdm_load_1d(const void* src, void* lds_dst, unsigned elems) {
  unsigned long long ga = (unsigned long long)(uintptr_t)src;
  unsigned lа = (unsigned)(uintptr_t)lds_dst;   // flat addr low 32 bits == LDS offset
  u32x4 g0;
  g0.x = 1u;                                    // count=1 (valid user descriptor)
  g0.y = lа;                                    // lds_addr
  g0.z = (unsigned)(ga & 0xFFFFFFFFu);          // global_addr[31:0]
  g0.w = (unsigned)((ga >> 32) & 0x1FFFFFFu) | (2u << 30);  // [56:32] | type=2
  i32x8 g1;
  g1[0] = (int)(2u << 16);                      // data_size = 4 bytes
  g1[1] = (int)((elems & 0xFFFFu) << 16);       // tensor_dim0[15:0]
  g1[2] = (int)((elems >> 16) | (1u << 16));    // tensor_dim0[31:16] | tensor_dim1=1
  g1[3] = (int)((elems & 0xFFFFu) << 16);       // tile_dim0
  g1[4] = 0;                                    // tile_dim1 = tile_dim2 = 0 (unused)
  g1[5] = (int)elems;                           // tensor_dim0_stride
  g1[6] = 0;
  g1[7] = 0;
  i32x4 z4 = {0, 0, 0, 0};
#if defined(__clang_major__) && (__clang_major__ >= 23)
  i32x8 z8 = {0, 0, 0, 0, 0, 0, 0, 0};
  __builtin_amdgcn_tensor_load_to_lds(g0, g1, z4, z4, z8, 0);
#else
  __builtin_amdgcn_tensor_load_to_lds(g0, g1, z4, z4, 0);
#endif
}
#else
#define HAVE_TDM 0
#endif

// ---------------------------------------------------------------------------
// fp32 WMMA GEMM with LDS-staged B: C[Nrows,NCOL] = A[Nrows,K] @ B[K,NCOL]
// One wave per 16x16 output tile; blockDim.x = 32 * (NCOL/16); one 16-row
// strip per block. B is staged in KC-row chunks, double-buffered via TDM.
// A 16x4 layout: lanes 0-15 -> M=lane, K=k,k+1 ; lanes 16-31 -> K=k+2,k+3.
// C layout: VGPR i -> M = i + 8*(lane>=16), N = lane&15.
// ---------------------------------------------------------------------------
template <int K, int NCOL, int KC>
__global__ void wmma_gemm_f32(const float* __restrict__ A,
                              const float* __restrict__ B,
                              float* __restrict__ C, int Nrows) {
  constexpr int CHUNK = KC * NCOL;   // elements per staged chunk (contiguous)
  constexpr int NC    = K / KC;
  __shared__ float lds[2 * CHUNK];

  const int wave = threadIdx.x >> 5;
  const int lane = threadIdx.x & 31;
  const int row0 = blockIdx.x * 16;
  const int m    = lane & 15;
  const int kh   = (lane >> 4) << 1;            // 0 or 2
  const int col  = (wave << 4) + (lane & 15);

  const float* arow = A + (size_t)(row0 + m) * K;

#if HAVE_TDM
  if (threadIdx.x == 0) tdm_load_1d(B, &lds[0], CHUNK);
#endif

  v8f c = {};
  for (int ch = 0; ch < NC; ++ch) {
    float* buf = &lds[(ch & 1) * CHUNK];
#if HAVE_TDM
    if (threadIdx.x == 0) __builtin_amdgcn_s_wait_tensorcnt(0);
    __syncthreads();                             // chunk ch visible to all waves
    if (ch + 1 < NC && threadIdx.x == 0)
      tdm_load_1d(B + (size_t)(ch + 1) * CHUNK, &lds[((ch + 1) & 1) * CHUNK], CHUNK);
#else
    // fallback: cooperative global -> LDS staging
    for (int i = threadIdx.x; i < CHUNK; i += blockDim.x)
      buf[i] = B[(size_t)ch * CHUNK + i];
    __syncthreads();
#endif
    const int kb = ch * KC;
#pragma unroll
    for (int k = 0; k < KC; k += 4) {
      v2f a, b;
      a.x = arow[kb + k + kh];
      a.y = arow[kb + k + kh + 1];
      b.x = buf[(k + kh) * NCOL + col];
      b.y = buf[(k + kh + 1) * NCOL + col];
      c = __builtin_amdgcn_wmma_f32_16x16x4_f32(false, a, false, b,
                                                (short)0, c, false, false);
    }
    __syncthreads();                             // all reads of buf done before reuse
  }

  const int mo = (lane >> 4) << 3;
  if (row0 + 16 <= Nrows) {
#pragma unroll
    for (int i = 0; i < 8; ++i)
      C[(size_t)(row0 + mo + i) * NCOL + col] = c[i];
  } else {
#pragma unroll
    for (int i = 0; i < 8; ++i) {
      int r = row0 + mo + i;
      if (r < Nrows) C[(size_t)r * NCOL + col] = c[i];
    }
  }
}

// ---------------------------------------------------------------------------
// Layer-1 attention coefficients: as/ad[n,h] = <h1[n,h,:], a_src/dst[h,:]>
// ---------------------------------------------------------------------------
__global__ void attn_coef1(const float* __restrict__ h1,
                           const float* __restrict__ a_src,
                           const float* __restrict__ a_dst,
                           float* __restrict__ as, float* __restrict__ ad, int N) {
  int t = blockIdx.x * blockDim.x + threadIdx.x;
  if (t >= N * H1) return;
  int n = t >> 3, h = t & 7;
  float ss = 0.f, dd = 0.f;
#pragma unroll
  for (int d = 0; d < D1; ++d) {
    float hv = h1[(size_t)n * C1 + h * D1 + d];
    ss += hv * a_src[h * D1 + d];
    dd += hv * a_dst[h * D1 + d];
  }
  as[t] = ss;
  ad[t] = dd;
}

__global__ void init1(unsigned* __restrict__ m1, float* __restrict__ s1,
                      float* __restrict__ acc1, int N) {
  int t = blockIdx.x * blockDim.x + threadIdx.x;
  if (t < N * C1) acc1[t] = 0.f;
  if (t < N * H1) { m1[t] = 0u; s1[t] = 0.f; }
}

__device__ __forceinline__ void edge_sd(int e, const int* __restrict__ ei,
                                        int E, int& s, int& d) {
  if (e < E) { s = ei[e]; d = ei[E + e]; }
  else       { s = d = e - E; }   // appended self loop
}

// per (edge, head)
__global__ void edge_max1(const int* __restrict__ ei, int E, int N,
                          const float* __restrict__ as, const float* __restrict__ ad,
                          unsigned* __restrict__ mkey) {
  int t = blockIdx.x * blockDim.x + threadIdx.x;
  int ET = E + N;
  if (t >= ET * H1) return;
  int e = t >> 3, h = t & 7, s, d;
  edge_sd(e, ei, E, s, d);
  float ev = lrelu(as[s * H1 + h] + ad[d * H1 + h]);
  atomicMaxF(&mkey[d * H1 + h], ev);
}

__global__ void edge_sum1(const int* __restrict__ ei, int E, int N,
                          const float* __restrict__ as, const float* __restrict__ ad,
                          const unsigned* __restrict__ mkey, float* __restrict__ ssum) {
  int t = blockIdx.x * blockDim.x + threadIdx.x;
  int ET = E + N;
  if (t >= ET * H1) return;
  int e = t >> 3, h = t & 7, s, d;
  edge_sd(e, ei, E, s, d);
  float ev = lrelu(as[s * H1 + h] + ad[d * H1 + h]);
  float ex = expf(ev - decodeMaxKey(mkey[d * H1 + h]));
  atomicAdd(&ssum[d * H1 + h], ex);
}

__global__ void edge_att1(const int* __restrict__ ei, int E, int N,
                          const float* __restrict__ as, const float* __restrict__ ad,
                          const unsigned* __restrict__ mkey, const float* __restrict__ ssum,
                          const float* __restrict__ h1, float* __restrict__ acc) {
  int t = blockIdx.x * blockDim.x + threadIdx.x;
  int ET = E + N;
  if (t >= ET * H1) return;
  int e = t >> 3, h = t & 7, s, d;
  edge_sd(e, ei, E, s, d);
  float ev = lrelu(as[s * H1 + h] + ad[d * H1 + h]);
  float alpha = expf(ev - decodeMaxKey(mkey[d * H1 + h])) / ssum[d * H1 + h];
#pragma unroll
  for (int k = 0; k < D1; ++k) {
    float v = h1[(size_t)s * C1 + h * D1 + k] * alpha;
    atomicAdd(&acc[(size_t)d * C1 + h * D1 + k], v);
  }
}

// h2in = elu(acc + b1)
__global__ void bias_elu(const float* __restrict__ acc, const float* __restrict__ b,
                         float* __restrict__ outp, int N) {
  int t = blockIdx.x * blockDim.x + threadIdx.x;
  if (t >= N * C1) return;
  float v = acc[t] + b[t & (C1 - 1)];
  outp[t] = v > 0.f ? v : (expf(v) - 1.f);
}

// layer-2: out init = b2 broadcast; alpha coefs (32-dot, single head); counters reset
__global__ void init2_alpha2(const float* __restrict__ hh2,
                             const float* __restrict__ a2s, const float* __restrict__ a2d,
                             float* __restrict__ as2, float* __restrict__ ad2,
                             unsigned* __restrict__ m2, float* __restrict__ s2,
                             const float* __restrict__ b2, float* __restrict__ outp, int N) {
  int t = blockIdx.x * blockDim.x + threadIdx.x;
  if (t < N * D2) outp[t] = b2[t & (D2 - 1)];
  if (t < N) {
    float ss = 0.f, dd = 0.f;
#pragma unroll
    for (int k = 0; k < D2; ++k) {
      float hv = hh2[(size_t)t * D2 + k];
      ss += hv * a2s[k];
      dd += hv * a2d[k];
    }
    as2[t] = ss; ad2[t] = dd;
    m2[t] = 0u;  s2[t] = 0.f;
  }
}

__global__ void edge_max2(const int* __restrict__ ei, int E, int N,
                          const float* __restrict__ as, const float* __restrict__ ad,
                          unsigned* __restrict__ mkey) {
  int e = blockIdx.x * blockDim.x + threadIdx.x;
  if (e >= E + N) return;
  int s, d; edge_sd(e, ei, E, s, d);
  atomicMaxF(&mkey[d], lrelu(as[s] + ad[d]));
}

__global__ void edge_sum2(const int* __restrict__ ei, int E, int N,
                          const float* __restrict__ as, const float* __restrict__ ad,
                          const unsigned* __restrict__ mkey, float* __restrict__ ssum) {
  int e = blockIdx.x * blockDim.x + threadIdx.x;
  if (e >= E + N) return;
  int s, d; edge_sd(e, ei, E, s, d);
  float ex = expf(lrelu(as[s] + ad[d]) - decodeMaxKey(mkey[d]));
  atomicAdd(&ssum[d], ex);
}

// per (edge, out-channel)
__global__ void edge_att2(const int* __restrict__ ei, int E, int N,
                          const float* __restrict__ as, const float* __restrict__ ad,
                          const unsigned* __restrict__ mkey, const float* __restrict__ ssum,
                          const float* __restrict__ hh2, float* __restrict__ outp) {
  int t = blockIdx.x * blockDim.x + threadIdx.x;
  int ET = E + N;
  if (t >= ET * D2) return;
  int e = t >> 5, j = t & (D2 - 1), s, d;
  edge_sd(e, ei, E, s, d);
  float alpha = expf(lrelu(as[s] + ad[d]) - decodeMaxKey(mkey[d])) / ssum[d];
  atomicAdd(&outp[(size_t)d * D2 + j], hh2[(size_t)s * D2 + j] * alpha);
}

// ---------------------------------------------------------------------------

static inline int cdiv(long long a, int b) { return (int)((a + b - 1) / b); }

extern "C" void kernel_launch(void* const* d_in, const int* in_sizes, int n_in,
                              void* d_out, int out_size, void* d_ws, size_t ws_size,
                              hipStream_t stream) {
  const float* x      = (const float*)d_in[0];
  const int*   ei     = (const int*)d_in[1];
  const float* W1     = (const float*)d_in[2];
  const float* a1_src = (const float*)d_in[3];
  const float* a1_dst = (const float*)d_in[4];
  const float* b1     = (const float*)d_in[5];
  const float* W2     = (const float*)d_in[6];
  const float* a2_src = (const float*)d_in[7];
  const float* a2_dst = (const float*)d_in[8];
  const float* b2     = (const float*)d_in[9];
  float* outp = (float*)d_out;

  const int N = in_sizes[0] / FIN;
  const int E = in_sizes[1] / 2;
  const int ET = E + N;

  // workspace carve-up (floats)
  float* p = (float*)d_ws;
  float*    h1   = p; p += (size_t)N * C1;
  float*    as1  = p; p += (size_t)N * H1;
  float*    ad1  = p; p += (size_t)N * H1;
  unsigned* m1   = (unsigned*)p; p += (size_t)N * H1;
  float*    s1   = p; p += (size_t)N * H1;
  float*    acc1 = p; p += (size_t)N * C1;
  float*    h2in = p; p += (size_t)N * C1;
  float*    hh2  = p; p += (size_t)N * D2;
  float*    as2  = p; p += (size_t)N;
  float*    ad2  = p; p += (size_t)N;
  unsigned* m2   = (unsigned*)p; p += (size_t)N;
  float*    s2   = p; p += (size_t)N;

  const int TB = 256;

  // ---- layer 1 ----
  wmma_gemm_f32<FIN, C1, 64><<<cdiv(N, 16), 32 * (C1 / 16), 0, stream>>>(x, W1, h1, N);
  attn_coef1<<<cdiv((long long)N * H1, TB), TB, 0, stream>>>(h1, a1_src, a1_dst, as1, ad1, N);
  init1<<<cdiv((long long)N * C1, TB), TB, 0, stream>>>(m1, s1, acc1, N);
  edge_max1<<<cdiv((long long)ET * H1, TB), TB, 0, stream>>>(ei, E, N, as1, ad1, m1);
  edge_sum1<<<cdiv((long long)ET * H1, TB), TB, 0, stream>>>(ei, E, N, as1, ad1, m1, s1);
  edge_att1<<<cdiv((long long)ET * H1, TB), TB, 0, stream>>>(ei, E, N, as1, ad1, m1, s1, h1, acc1);
  bias_elu<<<cdiv((long long)N * C1, TB), TB, 0, stream>>>(acc1, b1, h2in, N);

  // ---- layer 2 ----
  wmma_gemm_f32<C1, D2, 64><<<cdiv(N, 16), 32 * (D2 / 16), 0, stream>>>(h2in, W2, hh2, N);
  init2_alpha2<<<cdiv((long long)N * D2, TB), TB, 0, stream>>>(hh2, a2_src, a2_dst,
                                                               as2, ad2, m2, s2, b2, outp, N);
  edge_max2<<<cdiv(ET, TB), TB, 0, stream>>>(ei, E, N, as2, ad2, m2);
  edge_sum2<<<cdiv(ET, TB), TB, 0, stream>>>(ei, E, N, as2, ad2, m2, s2);
  edge_att2<<<cdiv((long long)ET * D2, TB), TB, 0, stream>>>(ei, E, N, as2, ad2, m2, s2, hh2, outp);
}